// UGT_85581518340119
// MI455X (gfx1250) — compile-verified
//
#include <hip/hip_runtime.h>
#include <hip/hip_bf16.h>
#include <math.h>

typedef __attribute__((ext_vector_type(2))) float v2f;
typedef __attribute__((ext_vector_type(8))) float v8f;

#define BATCH 8
#define CH 64
#define HH 128
#define WW 128
#define KANG 6
#define NBK (BATCH * KANG)          // 48
#define PIX (HH * WW)               // 16384
#define BN_EPS 1e-5f
#define PI_F 3.14159265358979323846f

// ---------------------------------------------------------------------------
// Kernel 1: AnglePredictor -> angles (B, K)
// ---------------------------------------------------------------------------
__global__ void angles_kernel(const float* __restrict__ x,
                              const float* __restrict__ w1, const float* __restrict__ b1,
                              const float* __restrict__ w2, const float* __restrict__ b2,
                              float* __restrict__ angles) {
    __shared__ float pooled[CH * 16];
    __shared__ float hbuf[32];
    const int b = blockIdx.x;

    // adaptive avg pool 4x4: pooled[c*16 + ph*4 + pw] = mean of 32x32 block
    for (int p = threadIdx.x; p < CH * 16; p += blockDim.x) {
        const int c = p >> 4, cell = p & 15, ph = cell >> 2, pw = cell & 3;
        const float* base = x + ((size_t)(b * CH + c) * HH + ph * 32) * WW + pw * 32;
        float s = 0.f;
        for (int i = 0; i < 32; ++i) {
            const float* r = base + i * WW;
            #pragma unroll 8
            for (int j = 0; j < 32; ++j) s += r[j];
        }
        pooled[p] = s * (1.0f / 1024.0f);
    }
    __syncthreads();

    if (threadIdx.x < 32) {
        const int j = threadIdx.x;
        float s = b1[j];
        const float* wr = w1 + j * (CH * 16);
        for (int i = 0; i < CH * 16; ++i) s += wr[i] * pooled[i];
        hbuf[j] = fmaxf(s, 0.f);
    }
    __syncthreads();

    if (threadIdx.x == 0) {
        float a[KANG];
        for (int k = 0; k < KANG; ++k) {
            float s = b2[k];
            const float* wr = w2 + k * 32;
            for (int i = 0; i < 32; ++i) s += wr[i] * hbuf[i];
            float ang = (float)k * 36.0f + s * 30.0f;   // base = linspace(0,180,6)
            a[k] = fminf(fmaxf(ang, 0.f), 180.f);
        }
        for (int i = 1; i < KANG; ++i) {                // sort ascending
            float v = a[i]; int j = i - 1;
            while (j >= 0 && a[j] > v) { a[j + 1] = a[j]; --j; }
            a[j + 1] = v;
        }
        for (int k = 0; k < KANG; ++k) angles[b * KANG + k] = a[k];
    }
}

// ---------------------------------------------------------------------------
// Kernel 2: Radon rotate (grid_sample, bilinear, zeros, AC=True) fused with
// sum over W.  One block per (n, h).  sig layout: (N=48, C=64, H=128)
// ---------------------------------------------------------------------------
__device__ __forceinline__ float fetch_img(const float* img, int y, int xq) {
    return (xq >= 0 && xq < WW && y >= 0 && y < HH) ? img[y * WW + xq] : 0.f;
}

__global__ void radon_kernel(const float* __restrict__ x,
                             const float* __restrict__ angles,
                             float* __restrict__ sig) {
    __shared__ float s_ix[WW];
    __shared__ float s_iy[WW];
    __shared__ float red[256];

    const int h = blockIdx.x;
    const int n = blockIdx.y;
    const int b = n / KANG;

    const float th = angles[n] * (PI_F / 180.0f);
    const float co = cosf(th), si = sinf(th);

    if (threadIdx.x < WW) {
        const int w = threadIdx.x;
        const float xc = -1.f + 2.f * (float)w * (1.0f / 127.0f);
        const float yc = -1.f + 2.f * (float)h * (1.0f / 127.0f);
        const float gx = xc * co + yc * si;
        const float gy = -xc * si + yc * co;
        s_ix[w] = (gx + 1.f) * 0.5f * (float)(WW - 1);
        s_iy[w] = (gy + 1.f) * 0.5f * (float)(HH - 1);
    }
    __syncthreads();

    const int c   = threadIdx.x >> 2;   // 0..63
    const int seg = threadIdx.x & 3;    // 0..3, each covers 32 w's
    const float* img = x + (size_t)(b * CH + c) * PIX;

    float acc = 0.f;
    const int w0 = seg * 32;
    #pragma unroll 4
    for (int w = w0; w < w0 + 32; ++w) {
        const float ix = s_ix[w], iy = s_iy[w];
        const float x0f = floorf(ix), y0f = floorf(iy);
        const int x0 = (int)x0f, y0 = (int)y0f;
        const float wx = ix - x0f, wy = iy - y0f;
        const float v00 = fetch_img(img, y0,     x0);
        const float v01 = fetch_img(img, y0,     x0 + 1);
        const float v10 = fetch_img(img, y0 + 1, x0);
        const float v11 = fetch_img(img, y0 + 1, x0 + 1);
        acc += v00 * (1.f - wx) * (1.f - wy) + v01 * wx * (1.f - wy)
             + v10 * (1.f - wx) * wy         + v11 * wx * wy;
    }
    red[threadIdx.x] = acc;
    __syncthreads();
    if (seg == 0) {
        sig[((size_t)n * CH + c) * HH + h] =
            red[c * 4] + red[c * 4 + 1] + red[c * 4 + 2] + red[c * 4 + 3];
    }
}

// ---------------------------------------------------------------------------
// Kernel 3: WaveletTransform1D (2 scales).  One 128-thread block per (n,c)
// row.  coeffs layout: (N=48, C=64, S=2, H=128)
// ---------------------------------------------------------------------------
__global__ void wavelet_kernel(const float* __restrict__ sig,
                               const float* __restrict__ wf0,
                               const float* __restrict__ wf1,
                               float* __restrict__ coeffs) {
    __shared__ float r[HH];
    __shared__ float cur[HH / 2];
    __shared__ float f1a[HH / 2];
    __shared__ float u1[HH / 2];

    const int row = blockIdx.x;          // n*64 + c
    const int t = threadIdx.x;
    r[t] = sig[(size_t)row * HH + t];
    __syncthreads();

    // scale 0: conv k=3 'same' (cross-correlation, zero pad)
    {
        float c0 = 0.f;
        #pragma unroll
        for (int j = 0; j < 3; ++j) {
            const int idx = t - 1 + j;
            const float v = (idx >= 0 && idx < HH) ? r[idx] : 0.f;
            c0 += v * wf0[j];
        }
        coeffs[((size_t)row * 2 + 0) * HH + t] = c0;
    }

    if (t < 64) cur[t] = 0.5f * (r[2 * t] + r[2 * t + 1]);   // avgpool
    __syncthreads();

    if (t < 64) {                                            // conv k=5 'same'
        float a = 0.f;
        #pragma unroll
        for (int j = 0; j < 5; ++j) {
            const int idx = t - 2 + j;
            const float v = (idx >= 0 && idx < 64) ? cur[idx] : 0.f;
            a += v * wf1[j];
        }
        f1a[t] = a;
    }
    __syncthreads();

    if (t < 64) {   // u1 = interp1d(avgpool(f1) [len 32] -> 64, AC=False)
        const float real = fminf(fmaxf(0.5f * (float)t - 0.25f, 0.f), 31.f);
        const int i0 = (int)floorf(real);
        const int i1 = min(i0 + 1, 31);
        const float wg = real - (float)i0;
        const float p0 = 0.5f * (f1a[2 * i0] + f1a[2 * i0 + 1]);
        const float p1 = 0.5f * (f1a[2 * i1] + f1a[2 * i1 + 1]);
        u1[t] = p0 * (1.f - wg) + p1 * wg;
    }
    __syncthreads();

    {   // c1 = interp1d(u1 [64] -> 128, AC=False)
        const float real = fminf(fmaxf(0.5f * (float)t - 0.25f, 0.f), 63.f);
        const int i0 = (int)floorf(real);
        const int i1 = min(i0 + 1, 63);
        const float wg = real - (float)i0;
        coeffs[((size_t)row * 2 + 1) * HH + t] = u1[i0] * (1.f - wg) + u1[i1] * wg;
    }
}

// ---------------------------------------------------------------------------
// Kernel 4: fused K-axis resize (AC=True) + 1x1-conv GEMM (WMMA f32 16x16x4)
// + bias + BatchNorm(eval) + ReLU.
//
// Per batch b:  out(64 x 16384) = fuse_w(64 x 128) @ Bmat(128 x 16384)
// Bmat[c2][p] is generated on the fly from coeffs via the reshape/view
// index algebra:  c = c2>>1, hUp = c2&1, h2 = p>>7, w = p&127,
//                 s = h2&1, h = hUp*64 + (h2>>1),
//                 pos = w*5/127, 2-tap interp over angle index k.
//
// Block: 256 threads = 8 waves; M=64 (4 m-tiles) x N=32 (2 n-tiles).
// ---------------------------------------------------------------------------
#define SA_STRIDE 130   // 128 + 2 pad: breaks bank conflicts on A-frag reads
#define NTILE 32

__global__ void fuse_wmma_kernel(const float* __restrict__ coeffs,
                                 const float* __restrict__ fuse_w,
                                 const float* __restrict__ fuse_b,
                                 const float* __restrict__ bn_gamma,
                                 const float* __restrict__ bn_beta,
                                 const float* __restrict__ bn_mean,
                                 const float* __restrict__ bn_var,
                                 float* __restrict__ out) {
    __shared__ float sA[CH * SA_STRIDE];      // fuse_w staged (64 x 128, padded)
    __shared__ float sB[128 * NTILE];         // interpolated B tile (K x N)

    const int b  = blockIdx.y;
    const int p0 = blockIdx.x * NTILE;        // 32 consecutive pixels, same h2
    const int h2 = p0 >> 7;
    const int wbase = p0 & (WW - 1);
    const int s  = h2 & 1;
    const int hlow = h2 >> 1;

    // --- stage A (fuse_w) ---
    for (int idx = threadIdx.x; idx < CH * 128; idx += blockDim.x) {
        const int o = idx >> 7, c2 = idx & 127;
        sA[o * SA_STRIDE + c2] = fuse_w[o * 128 + c2];
    }

    // --- stage B tile: resize along K (align_corners=True) ---
    for (int idx = threadIdx.x; idx < 128 * NTILE; idx += blockDim.x) {
        const int c2 = idx >> 5;              // 0..127
        const int j  = idx & 31;              // pixel within tile
        const int w  = wbase + j;
        const float pos = (float)w * ((float)(KANG - 1) / (float)(WW - 1));
        const int k0 = (int)floorf(pos);
        const int k1 = min(k0 + 1, KANG - 1);
        const float wk = pos - (float)k0;
        const int c = c2 >> 1;
        const int h = (c2 & 1) * 64 + hlow;
        const size_t i0 = (((size_t)(b * KANG + k0) * CH + c) * 2 + s) * HH + h;
        const size_t i1 = (((size_t)(b * KANG + k1) * CH + c) * 2 + s) * HH + h;
        sB[c2 * NTILE + j] = coeffs[i0] * (1.f - wk) + coeffs[i1] * wk;
    }
    __syncthreads();

    // --- WMMA main loop ---
    const int wv   = threadIdx.x >> 5;        // wave 0..7
    const int m    = wv & 3;                  // m-tile (output-channel tile)
    const int nn   = wv >> 2;                 // n-tile (pixel tile)
    const int lane = threadIdx.x & 31;
    const int laneLow = lane & 15;
    const int laneHi  = lane >> 4;            // 0: K{0,1}, 1: K{2,3}

    const int arow = m * 16 + laneLow;        // output channel this lane feeds A
    const int bcol = nn * 16 + laneLow;       // pixel column this lane feeds B

    v8f acc = {};
    #pragma unroll
    for (int kk = 0; kk < 32; ++kk) {         // K = 128 in steps of 4
        const int kb = kk * 4 + laneHi * 2;
        v2f a, bf;
        a.x  = sA[arow * SA_STRIDE + kb];
        a.y  = sA[arow * SA_STRIDE + kb + 1];
        bf.x = sB[kb * NTILE + bcol];
        bf.y = sB[(kb + 1) * NTILE + bcol];
        acc = __builtin_amdgcn_wmma_f32_16x16x4_f32(
                  /*neg_a=*/false, a, /*neg_b=*/false, bf,
                  /*c_mod=*/(short)0, acc, /*reuse_a=*/false, /*reuse_b=*/false);
    }

    // --- epilogue: bias + BN(eval) + ReLU, D-matrix layout:
    //     lane<16: M=v, N=lane ; lane>=16: M=8+v, N=lane-16
    const int ncol = bcol;                    // == nn*16 + (lane mod 16)
    #pragma unroll
    for (int v = 0; v < 8; ++v) {
        const int o = m * 16 + laneHi * 8 + v;
        const float sc = bn_gamma[o] * __frsqrt_rn(bn_var[o] + BN_EPS);
        float val = (acc[v] + fuse_b[o] - bn_mean[o]) * sc + bn_beta[o];
        val = fmaxf(val, 0.f);
        out[(size_t)b * CH * PIX + (size_t)o * PIX + (p0 & ~(NTILE - 1)) + ncol] = val;
    }
}

// ---------------------------------------------------------------------------
extern "C" void kernel_launch(void* const* d_in, const int* in_sizes, int n_in,
                              void* d_out, int out_size, void* d_ws, size_t ws_size,
                              hipStream_t stream) {
    const float* x        = (const float*)d_in[0];
    const float* lin1_w   = (const float*)d_in[1];
    const float* lin1_b   = (const float*)d_in[2];
    const float* lin2_w   = (const float*)d_in[3];
    const float* lin2_b   = (const float*)d_in[4];
    const float* wf0      = (const float*)d_in[5];
    const float* wf1      = (const float*)d_in[6];
    const float* fuse_w   = (const float*)d_in[7];
    const float* fuse_b   = (const float*)d_in[8];
    const float* bn_gamma = (const float*)d_in[9];
    const float* bn_beta  = (const float*)d_in[10];
    const float* bn_mean  = (const float*)d_in[11];
    const float* bn_var   = (const float*)d_in[12];
    float* out = (float*)d_out;

    float* ws      = (float*)d_ws;
    float* angles  = ws;                                   // 48
    float* sig     = ws + 64;                              // 48*64*128 = 393216
    float* coeffs  = ws + 64 + NBK * CH * HH;              // 48*64*2*128 = 786432

    angles_kernel<<<dim3(BATCH), dim3(256), 0, stream>>>(
        x, lin1_w, lin1_b, lin2_w, lin2_b, angles);

    radon_kernel<<<dim3(HH, NBK), dim3(256), 0, stream>>>(x, angles, sig);

    wavelet_kernel<<<dim3(NBK * CH), dim3(HH), 0, stream>>>(sig, wf0, wf1, coeffs);

    fuse_wmma_kernel<<<dim3(PIX / NTILE, BATCH), dim3(256), 0, stream>>>(
        coeffs, fuse_w, fuse_b, bn_gamma, bn_beta, bn_mean, bn_var, out);
}